// ModifiedMarginLoss_652835029338
// MI455X (gfx1250) — compile-verified
//
#include <hip/hip_runtime.h>
#include <hip/hip_bf16.h>

typedef float v2f __attribute__((ext_vector_type(2)));
typedef float v8f __attribute__((ext_vector_type(8)));

#define D 128           // feature dim (K)
#define WAVES 8         // waves per block
#define ALPHA_C 0.1f
#define EPS_C 1e-5f
#define JMAX 0x7FFFFFFF

// Load one 16x16-tile's B fragments (and its column target) into registers.
// B 4x16 fp32 layout mirrors A: lane holds N=l16, VGPR0/1 = K = koff+0/1.
#define LOAD_TILE(buf, tt, tc)                                                \
    do {                                                                      \
        const float* brow_ = X + (size_t)((tt) * 16 + l16) * D;               \
        (tc) = tgt[(tt) * 16 + l16];                                          \
        _Pragma("unroll")                                                     \
        for (int kk_ = 0; kk_ < 32; ++kk_)                                    \
            buf[kk_] = *(const v2f*)(brow_ + 4 * kk_ + koff);                 \
    } while (0)

// 32-step WMMA chain over K=128 + fused masked-margin epilogue.
#define COMPUTE_TILE(buf, tt, tc)                                             \
    do {                                                                      \
        const int col_ = (tt) * 16 + l16;                                     \
        v8f c_ = {};                                                          \
        _Pragma("unroll")                                                     \
        for (int kk_ = 0; kk_ < 32; ++kk_)                                    \
            c_ = __builtin_amdgcn_wmma_f32_16x16x4_f32(                       \
                     false, a[kk_], false, buf[kk_], (short)0, c_,            \
                     false, false);                                           \
        _Pragma("unroll")                                                     \
        for (int v_ = 0; v_ < 8; ++v_) {                                      \
            const float s_ = c_[v_];                                          \
            const bool pos_ = (trow[v_] == (tc));                             \
            const float pv_ = pos_ ? fmaxf(ALPHA_C + beta - s_, 0.f) : 0.f;   \
            const float nv_ = pos_ ? 0.f : fmaxf(s_ - beta + ALPHA_C, 0.f);   \
            psum[v_] += pv_;                  nsum[v_] += nv_;                 \
            pmax[v_] = fmaxf(pmax[v_], pv_);  nmax[v_] = fmaxf(nmax[v_], nv_);\
            if (pos_ && col_ < pfj[v_]) { pfj[v_] = col_; pfv[v_] = pv_; }    \
            if (!pos_ && col_ < nfj[v_]) { nfj[v_] = col_; nfv[v_] = nv_; }   \
        }                                                                     \
    } while (0)

// Each block handles a 16-row band of sim = X * X^T using V_WMMA_F32_16X16X4_F32,
// fusing the masked relu-margin row reductions. Writes one partial per block to ws.
__global__ void __launch_bounds__(32 * WAVES)
margin_main(const float* __restrict__ X, const int* __restrict__ tgt,
            const float* __restrict__ beta_p, float* __restrict__ ws, int n) {
    const int wave = threadIdx.x >> 5;
    const int lane = threadIdx.x & 31;
    const int l16  = lane & 15;
    const int hi   = lane >> 4;       // 0: lanes 0-15, 1: lanes 16-31
    const int koff = hi * 2;          // fp32 WMMA K split across lane halves
    const int rowb = blockIdx.x * 16;
    const float beta = beta_p[0];

    // ---- Preload A fragments for this 16-row band (reused for every tile) ----
    // A 16x4 fp32 layout: lane L holds M=L%16; VGPR0/1 = K = koff+0 / koff+1.
    v2f a[32];
    const float* arow = X + (size_t)(rowb + l16) * D;
#pragma unroll
    for (int kk = 0; kk < 32; ++kk)
        a[kk] = *(const v2f*)(arow + 4 * kk + koff);

    // Row targets for the 8 rows this lane's C fragments cover (M = v + hi*8).
    int trow[8];
#pragma unroll
    for (int v = 0; v < 8; ++v)
        trow[v] = tgt[rowb + v + hi * 8];

    float psum[8], pmax[8], pfv[8], nsum[8], nmax[8], nfv[8];
    int pfj[8], nfj[8];
#pragma unroll
    for (int v = 0; v < 8; ++v) {
        psum[v] = 0.f; pmax[v] = 0.f; pfv[v] = 0.f; pfj[v] = JMAX;
        nsum[v] = 0.f; nmax[v] = 0.f; nfv[v] = 0.f; nfj[v] = JMAX;
    }

    // ---- Software-pipelined tile loop: double-buffered B in registers ----
    const int ntiles = n >> 4;
    v2f b0[32], b1[32];
    int tc0 = 0, tc1 = 0;

    if (wave < ntiles) LOAD_TILE(b0, wave, tc0);
    for (int t = wave; t < ntiles; t += 2 * WAVES) {
        const int t1 = t + WAVES;
        const int t2 = t + 2 * WAVES;
        if (t1 < ntiles) LOAD_TILE(b1, t1, tc1);   // prefetch next tile
        COMPUTE_TILE(b0, t, tc0);
        if (t2 < ntiles) LOAD_TILE(b0, t2, tc0);   // prefetch next-next
        if (t1 < ntiles) COMPUTE_TILE(b1, t1, tc1);
    }

    // ---- Reduce across the 16 columns held by each 16-lane half (wave32) ----
#pragma unroll
    for (int v = 0; v < 8; ++v) {
#pragma unroll
        for (int o = 1; o < 16; o <<= 1) {
            psum[v] += __shfl_xor(psum[v], o, 32);
            nsum[v] += __shfl_xor(nsum[v], o, 32);
            pmax[v] = fmaxf(pmax[v], __shfl_xor(pmax[v], o, 32));
            nmax[v] = fmaxf(nmax[v], __shfl_xor(nmax[v], o, 32));
            int   oj = __shfl_xor(pfj[v], o, 32);
            float ov = __shfl_xor(pfv[v], o, 32);
            if (oj < pfj[v]) { pfj[v] = oj; pfv[v] = ov; }
            oj = __shfl_xor(nfj[v], o, 32);
            ov = __shfl_xor(nfv[v], o, 32);
            if (oj < nfj[v]) { nfj[v] = oj; nfv[v] = ov; }
        }
    }

    // ---- Cross-wave combine via LDS (fixed order -> deterministic) ----
    __shared__ float fbuf[WAVES][16][6];
    __shared__ int   ibuf[WAVES][16][2];
    __shared__ float rowc[16];

    if (l16 == 0) {  // lanes 0 and 16 hold the reduced values
#pragma unroll
        for (int v = 0; v < 8; ++v) {
            const int r = v + hi * 8;
            fbuf[wave][r][0] = psum[v];
            fbuf[wave][r][1] = pmax[v];
            fbuf[wave][r][2] = pfv[v];
            fbuf[wave][r][3] = nsum[v];
            fbuf[wave][r][4] = nmax[v];
            fbuf[wave][r][5] = nfv[v];
            ibuf[wave][r][0] = pfj[v];
            ibuf[wave][r][1] = nfj[v];
        }
    }
    __syncthreads();

    if (threadIdx.x < 16) {
        const int r = threadIdx.x;
        float ps = 0.f, pm = 0.f, pv = 0.f, ns = 0.f, nm = 0.f, nv = 0.f;
        int pj = JMAX, nj = JMAX;
        for (int w = 0; w < WAVES; ++w) {
            ps += fbuf[w][r][0];
            pm = fmaxf(pm, fbuf[w][r][1]);
            if (ibuf[w][r][0] < pj) { pj = ibuf[w][r][0]; pv = fbuf[w][r][2]; }
            ns += fbuf[w][r][3];
            nm = fmaxf(nm, fbuf[w][r][4]);
            if (ibuf[w][r][1] < nj) { nj = ibuf[w][r][1]; nv = fbuf[w][r][5]; }
        }
        float contrib = 0.f;
        if (pm / (ps + EPS_C) > 0.5f) contrib += pv;
        if (nm / (ns + EPS_C) > 0.5f) contrib += nv;
        rowc[r] = contrib;
    }
    __syncthreads();

    if (threadIdx.x == 0) {
        float s = 0.f;
        for (int r = 0; r < 16; ++r) s += rowc[r];
        ws[blockIdx.x] = s;
    }
}

// Deterministic fixed-order tree reduction of the per-block partials.
__global__ void __launch_bounds__(256)
margin_reduce(const float* __restrict__ ws, float* __restrict__ out, int nblocks) {
    __shared__ float s[256];
    const int t = threadIdx.x;
    float acc = 0.f;
    for (int i = t; i < nblocks; i += 256) acc += ws[i];
    s[t] = acc;
    __syncthreads();
    for (int o = 128; o > 0; o >>= 1) {
        if (t < o) s[t] += s[t + o];
        __syncthreads();
    }
    if (t == 0) out[0] = s[0];
}

extern "C" void kernel_launch(void* const* d_in, const int* in_sizes, int n_in,
                              void* d_out, int out_size, void* d_ws, size_t ws_size,
                              hipStream_t stream) {
    const float* X      = (const float*)d_in[0];
    const int*   tgt    = (const int*)d_in[1];
    const float* beta_p = (const float*)d_in[2];
    float*       out    = (float*)d_out;
    float*       ws     = (float*)d_ws;

    const int n = in_sizes[1];        // number of samples (targets count)
    const int nblocks = n / 16;       // 16-row band per block

    margin_main<<<nblocks, 32 * WAVES, 0, stream>>>(X, tgt, beta_p, ws, n);
    margin_reduce<<<1, 256, 0, stream>>>(ws, out, nblocks);
}